// SEBlock_62173946577551
// MI455X (gfx1250) — compile-verified
//
#include <hip/hip_runtime.h>
#include <hip/hip_bf16.h>
#include <math.h>

// SE block: x[32,512,64,64] fp32; w1[32,512]; w2[512,32]
//   y = mean(x, HW)                      [32,512]
//   h = relu(y @ w1^T)                   [32,32]
//   s = sigmoid(h @ w2^T)                [32,512]
//   out = x * s (broadcast over HW)
//
// Memory-bound: ~805 MB traffic -> ~35us floor @ 23.3 TB/s. GEMMs done with
// exact-fp32 V_WMMA_F32_16X16X4_F32 (CDNA5 matrix path, no precision loss).

typedef __attribute__((ext_vector_type(2))) float v2f;
typedef __attribute__((ext_vector_type(8))) float v8f;

#define NPLANE 16384      // 32 batches * 512 channels
#define PLANE_ELEMS 4096  // 64*64

// ---------------- Kernel 1: global average pool (one block per (b,c) plane)
__global__ void se_pool(const float* __restrict__ x, float* __restrict__ y) {
    __shared__ float red[256];
    const int plane = blockIdx.x;
    const float4* xp = (const float4*)(x + (size_t)plane * PLANE_ELEMS);
    float s = 0.0f;
#pragma unroll
    for (int i = 0; i < 4; ++i) {
        float4 v = xp[threadIdx.x + i * 256];
        s += (v.x + v.y) + (v.z + v.w);
    }
    red[threadIdx.x] = s;
    __syncthreads();
#pragma unroll
    for (int off = 128; off > 0; off >>= 1) {
        if (threadIdx.x < off) red[threadIdx.x] += red[threadIdx.x + off];
        __syncthreads();
    }
    if (threadIdx.x == 0) y[plane] = red[0] * (1.0f / (float)PLANE_ELEMS);
}

// ---------------- Kernel 2: excitation MLP via fp32 WMMA (single block, 8 waves)
// h = relu(y @ w1^T) [32,32]; s = sigmoid(h @ w2^T) [32,512]
__global__ void se_mlp(const float* __restrict__ y,   // [32,512]
                       const float* __restrict__ w1,  // [32,512]
                       const float* __restrict__ w2,  // [512,32]
                       float* __restrict__ s) {       // [32,512]
    __shared__ float h[32 * 32];

    const int lane  = threadIdx.x & 31;
    const int wave  = threadIdx.x >> 5;
    const int r     = lane & 15;   // row within 16 (A: M row, B: N col)
    const int khalf = lane >> 4;   // 0 => K{0,1}, 1 => K{2,3}

    // ---- Stage A: 4 tiles of h = relu(y @ w1^T), K = 512
    if (wave < 4) {
        const int mt = wave >> 1;
        const int nt = wave & 1;
        v8f c = {};
        const float* arow = y  + (size_t)(mt * 16 + r) * 512 + 2 * khalf;
        const float* brow = w1 + (size_t)(nt * 16 + r) * 512 + 2 * khalf; // B[k][n]=w1[n][k]
        for (int k = 0; k < 512; k += 4) {
            v2f a = *(const v2f*)(arow + k);
            v2f b = *(const v2f*)(brow + k);
            c = __builtin_amdgcn_wmma_f32_16x16x4_f32(
                    false, a, false, b, (short)0, c, false, false);
        }
#pragma unroll
        for (int i = 0; i < 8; ++i) {
            const int row = mt * 16 + i + 8 * khalf;  // C/D layout: VGPR i -> M=i / M=i+8
            const int col = nt * 16 + r;
            h[row * 32 + col] = fmaxf(c[i], 0.0f);    // ReLU
        }
    }
    __syncthreads();

    // ---- Stage B: 64 tiles (2 x 32) of s = sigmoid(h @ w2^T), K = 32
#pragma unroll
    for (int tt = 0; tt < 8; ++tt) {
        const int tile = wave * 8 + tt;
        const int mt = tile & 1;
        const int nt = tile >> 1;
        v8f c = {};
        const float* arow = h  + (mt * 16 + r) * 32 + 2 * khalf;
        const float* brow = w2 + (size_t)(nt * 16 + r) * 32 + 2 * khalf;  // B[k][n]=w2[n][k]
#pragma unroll
        for (int k = 0; k < 32; k += 4) {
            v2f a = *(const v2f*)(arow + k);
            v2f b = *(const v2f*)(brow + k);
            c = __builtin_amdgcn_wmma_f32_16x16x4_f32(
                    false, a, false, b, (short)0, c, false, false);
        }
#pragma unroll
        for (int i = 0; i < 8; ++i) {
            const int row = mt * 16 + i + 8 * khalf;
            const int col = nt * 16 + r;
            s[row * 512 + col] = 1.0f / (1.0f + expf(-c[i]));  // sigmoid
        }
    }
}

// ---------------- Kernel 3: channel-wise scale (one block per plane)
__global__ void se_scale(const float* __restrict__ x, const float* __restrict__ s,
                         float* __restrict__ out) {
    const int plane = blockIdx.x;
    const float g = s[plane];  // uniform per block -> scalar load
    const float4* xp = (const float4*)(x + (size_t)plane * PLANE_ELEMS);
    float4* op = (float4*)(out + (size_t)plane * PLANE_ELEMS);
#pragma unroll
    for (int i = 0; i < 4; ++i) {
        float4 v = xp[threadIdx.x + i * 256];
        v.x *= g; v.y *= g; v.z *= g; v.w *= g;
        op[threadIdx.x + i * 256] = v;
    }
}

extern "C" void kernel_launch(void* const* d_in, const int* in_sizes, int n_in,
                              void* d_out, int out_size, void* d_ws, size_t ws_size,
                              hipStream_t stream) {
    const float* x  = (const float*)d_in[0];  // [32,512,64,64]
    const float* w1 = (const float*)d_in[1];  // [32,512]
    const float* w2 = (const float*)d_in[2];  // [512,32]
    float* out = (float*)d_out;

    float* y = (float*)d_ws;      // [32,512] pooled means
    float* s = y + NPLANE;        // [32,512] sigmoid scales (128 KB total ws)

    se_pool <<<NPLANE, 256, 0, stream>>>(x, y);
    se_mlp  <<<1,      256, 0, stream>>>(y, w1, w2, s);
    se_scale<<<NPLANE, 256, 0, stream>>>(x, s, out);
}